// GatLayer_35296041238626
// MI455X (gfx1250) — compile-verified
//
#include <hip/hip_runtime.h>
#include <math.h>

typedef __attribute__((ext_vector_type(2))) float v2f;
typedef __attribute__((ext_vector_type(8))) float v8f;

#define F        128     // F_IN == H*F_OUT == 128
#define NHEAD    8
#define NEG_SLOPE 0.2f
#define LN_EPS   1e-5f

__device__ __forceinline__ float leaky(float v) { return v > 0.f ? v : NEG_SLOPE * v; }

// ---------------------------------------------------------------------------
// K0: initialize accumulators (d_out / d_ws are poisoned by the harness)
// ---------------------------------------------------------------------------
__global__ void init_bufs(float* __restrict__ maxv, float* __restrict__ sumv,
                          float* __restrict__ out, int n) {
    int i = blockIdx.x * blockDim.x + threadIdx.x;
    if (i < n * NHEAD) { maxv[i] = -INFINITY; sumv[i] = 0.f; }
    if (i < n * F)     { out[i] = 0.f; }
}

// ---------------------------------------------------------------------------
// K0b: repack W (128x128) into per-wave WMMA B-fragment order so the GEMM
//      inner loop does ONE coalesced b64 load per v_wmma:
//      Wf[((head*32 + kk)*32 + lane)*2 + {0,1}] = W[kb][head*16 + m], W[kb+1][...]
//      with m = lane&15, kp = lane>>4, kb = kk*4 + 2*kp   (64 KB total)
// ---------------------------------------------------------------------------
__global__ void pack_w(const float* __restrict__ W, float* __restrict__ Wf) {
    int i = blockIdx.x * blockDim.x + threadIdx.x;    // head*1024 + kk*32 + lane
    if (i >= NHEAD * 32 * 32) return;
    const int lane = i & 31, kk = (i >> 5) & 31, head = i >> 10;
    const int m  = lane & 15;
    const int kp = lane >> 4;
    const int kb = kk * 4 + 2 * kp;
    Wf[i * 2 + 0] = W[kb * F + head * 16 + m];
    Wf[i * 2 + 1] = W[(kb + 1) * F + head * 16 + m];
}

// ---------------------------------------------------------------------------
// K1: xp = x @ W (fp32, exact) via V_WMMA_F32_16X16X4_F32
//   block = 256 threads = 8 waves; x strip staged in LDS through the gfx1250
//   async-to-LDS engine (global_load_async_to_lds_b128, ASYNCcnt-tracked);
//   wave w computes the 16x16 tile for head w. K=128 -> 32 chained wmma ops.
// ---------------------------------------------------------------------------
__global__ void __launch_bounds__(256) gemm_xw(const float* __restrict__ x,
                                               const float* __restrict__ Wf,
                                               float* __restrict__ xp, int n) {
    __shared__ float xs[16 * F];          // 8 KB strip of x
    const int wave = threadIdx.x >> 5;
    const int lane = threadIdx.x & 31;
    const int row0 = blockIdx.x * 16;

    // async copy: 16 rows x 128 cols = 512 float4, 2 per thread, no VGPR bounce
    {
        const float4* xg = (const float4*)(x + (size_t)row0 * F);
        const unsigned lds0 = (unsigned)(uintptr_t)&xs[0];
        const int nv = min(512, (n - row0) * (F / 4));
        for (int i = threadIdx.x; i < nv; i += 256) {
            const float4* gp = xg + i;
            unsigned laddr = lds0 + (unsigned)i * 16u;
            asm volatile("global_load_async_to_lds_b128 %0, %1, off"
                         :: "v"(laddr), "v"(gp) : "memory");
        }
        asm volatile("s_wait_asynccnt 0x0" ::: "memory");
    }
    __syncthreads();

    const int m  = lane & 15;             // A row within tile
    // B fragments: fragment-order repacked, one coalesced b64 per iteration
    const float* wfb = Wf + ((size_t)(wave * 32) * 32 + lane) * 2;
    const float* xrow = xs + m * F + 2 * (lane >> 4);

    v8f c = {};
#pragma unroll
    for (int kk = 0; kk < F / 4; ++kk) {
        v2f a, b;
        a.x = xrow[kk * 4];
        a.y = xrow[kk * 4 + 1];
        b   = *(const v2f*)(wfb + (size_t)kk * 64);
        // 8 args: (neg_a, A, neg_b, B, c_mod, C, reuse_a, reuse_b)
        c = __builtin_amdgcn_wmma_f32_16x16x4_f32(false, a, false, b,
                                                  (short)0, c, false, false);
    }

    const int rg = lane >> 4;
#pragma unroll
    for (int v = 0; v < 8; ++v) {
        const int r = row0 + v + 8 * rg;
        if (r < n) xp[(size_t)r * F + wave * 16 + m] = c[v];
    }
}

// ---------------------------------------------------------------------------
// K2: a_src[n,h] = <xp[n,h,:], att_src[h,:]> ; same for a_dst
// ---------------------------------------------------------------------------
__global__ void attn_scores(const float* __restrict__ xp,
                            const float* __restrict__ att_src,
                            const float* __restrict__ att_dst,
                            float* __restrict__ a_src, float* __restrict__ a_dst,
                            int n) {
    int idx = blockIdx.x * blockDim.x + threadIdx.x;   // node*NHEAD + h
    if (idx >= n * NHEAD) return;
    const int h = idx & (NHEAD - 1);
    const float* v  = xp + (size_t)(idx >> 3) * F + h * 16;
    const float* as = att_src + h * 16;
    const float* ad = att_dst + h * 16;
    float s = 0.f, d = 0.f;
#pragma unroll
    for (int f = 0; f < 16; ++f) { float t = v[f]; s += t * as[f]; d += t * ad[f]; }
    a_src[idx] = s;
    a_dst[idx] = d;
}

__device__ __forceinline__ void edge_nodes(const int* __restrict__ ei, int E,
                                           int e, int& s, int& d) {
    if (e < E) { s = ei[e]; d = ei[E + e]; } else { s = d = e - E; }
}

// ---------------------------------------------------------------------------
// K3: per-destination segment max (thread per edge, float4 gathers, 8 atomics)
// ---------------------------------------------------------------------------
__global__ void edge_max(const int* __restrict__ ei, int E, int ET,
                         const float* __restrict__ a_src,
                         const float* __restrict__ a_dst,
                         float* __restrict__ maxv) {
    int e = blockIdx.x * blockDim.x + threadIdx.x;
    if (e >= ET) return;
    int s, d; edge_nodes(ei, E, e, s, d);
    const float4* asp = (const float4*)(a_src + (size_t)s * NHEAD);
    const float4* adp = (const float4*)(a_dst + (size_t)d * NHEAD);
    const float4 as0 = asp[0], as1 = asp[1];
    const float4 ad0 = adp[0], ad1 = adp[1];
    float* mv = maxv + (size_t)d * NHEAD;
    atomicMax(mv + 0, leaky(as0.x + ad0.x));
    atomicMax(mv + 1, leaky(as0.y + ad0.y));
    atomicMax(mv + 2, leaky(as0.z + ad0.z));
    atomicMax(mv + 3, leaky(as0.w + ad0.w));
    atomicMax(mv + 4, leaky(as1.x + ad1.x));
    atomicMax(mv + 5, leaky(as1.y + ad1.y));
    atomicMax(mv + 6, leaky(as1.z + ad1.z));
    atomicMax(mv + 7, leaky(as1.w + ad1.w));
}

// ---------------------------------------------------------------------------
// K4: segment sum of exp(e - max) (thread per edge)
// ---------------------------------------------------------------------------
__global__ void edge_sum(const int* __restrict__ ei, int E, int ET,
                         const float* __restrict__ a_src,
                         const float* __restrict__ a_dst,
                         const float* __restrict__ maxv,
                         float* __restrict__ sumv) {
    int e = blockIdx.x * blockDim.x + threadIdx.x;
    if (e >= ET) return;
    int s, d; edge_nodes(ei, E, e, s, d);
    const float4* asp = (const float4*)(a_src + (size_t)s * NHEAD);
    const float4* adp = (const float4*)(a_dst + (size_t)d * NHEAD);
    const float4* mvp = (const float4*)(maxv + (size_t)d * NHEAD);
    const float4 as0 = asp[0], as1 = asp[1];
    const float4 ad0 = adp[0], ad1 = adp[1];
    const float4 m0 = mvp[0], m1 = mvp[1];
    float* sv = sumv + (size_t)d * NHEAD;
    atomicAdd(sv + 0, __expf(leaky(as0.x + ad0.x) - m0.x));
    atomicAdd(sv + 1, __expf(leaky(as0.y + ad0.y) - m0.y));
    atomicAdd(sv + 2, __expf(leaky(as0.z + ad0.z) - m0.z));
    atomicAdd(sv + 3, __expf(leaky(as0.w + ad0.w) - m0.w));
    atomicAdd(sv + 4, __expf(leaky(as1.x + ad1.x) - m1.x));
    atomicAdd(sv + 5, __expf(leaky(as1.y + ad1.y) - m1.y));
    atomicAdd(sv + 6, __expf(leaky(as1.z + ad1.z) - m1.z));
    atomicAdd(sv + 7, __expf(leaky(as1.w + ad1.w) - m1.w));
}

// ---------------------------------------------------------------------------
// K5: scatter  out[dst] += alpha * xp[src]   (one wave32 per edge; lanes 0-7
//     compute the 8 per-head alphas (EXEC-masked gathers), shfl-broadcast,
//     each lane then handles 4 contiguous features: float4 gather + 4 atomics.
//     Working set (xp 25.6MB + out 25.6MB + scalars) is L2-resident.
// ---------------------------------------------------------------------------
__global__ void __launch_bounds__(256) edge_scatter(
        const int* __restrict__ ei, int E, int ET,
        const float* __restrict__ a_src, const float* __restrict__ a_dst,
        const float* __restrict__ maxv, const float* __restrict__ sumv,
        const float* __restrict__ xp, float* __restrict__ out) {
    const int e = blockIdx.x * 8 + (threadIdx.x >> 5);
    if (e >= ET) return;
    const int t = threadIdx.x & 31;
    int s, d; edge_nodes(ei, E, e, s, d);

    float av = 0.f;
    if (t < NHEAD) {                            // only 8 lanes issue these gathers
        const float v = leaky(a_src[(size_t)s * NHEAD + t] +
                              a_dst[(size_t)d * NHEAD + t]);
        const float w = __expf(v - maxv[(size_t)d * NHEAD + t]);
        const float z = sumv[(size_t)d * NHEAD + t];
        av = w / (z > 0.f ? z : 1.f);
    }
    const float alpha = __shfl(av, t >> 2, 32); // head = (t*4)/16

    const float4 xv = *(const float4*)(xp + (size_t)s * F + t * 4);
    float* o = out + (size_t)d * F + t * 4;
    atomicAdd(o + 0, alpha * xv.x);
    atomicAdd(o + 1, alpha * xv.y);
    atomicAdd(o + 2, alpha * xv.z);
    atomicAdd(o + 3, alpha * xv.w);
}

// ---------------------------------------------------------------------------
// K6: bias + LayerNorm(128) in place on d_out; one wave per node,
//     float4 per lane + wave32 butterfly reduction.
// ---------------------------------------------------------------------------
__global__ void __launch_bounds__(256) layernorm(float* __restrict__ out,
                                                 const float* __restrict__ bias,
                                                 const float* __restrict__ gamma,
                                                 const float* __restrict__ beta,
                                                 int n) {
    const int node = blockIdx.x * 8 + (threadIdx.x >> 5);
    if (node >= n) return;
    const int t = threadIdx.x & 31;
    float4 v = *(float4*)(out + (size_t)node * F + t * 4);
    const float4 b = *(const float4*)(bias + t * 4);
    v.x += b.x; v.y += b.y; v.z += b.z; v.w += b.w;

    float s  = v.x + v.y + v.z + v.w;
    float sq = v.x * v.x + v.y * v.y + v.z * v.z + v.w * v.w;
#pragma unroll
    for (int off = 16; off > 0; off >>= 1) {
        s  += __shfl_xor(s,  off, 32);
        sq += __shfl_xor(sq, off, 32);
    }
    const float mu  = s * (1.f / F);
    const float var = sq * (1.f / F) - mu * mu;
    const float rs  = rsqrtf(var + LN_EPS);

    const float4 g  = *(const float4*)(gamma + t * 4);
    const float4 be = *(const float4*)(beta + t * 4);
    v.x = (v.x - mu) * rs * g.x + be.x;
    v.y = (v.y - mu) * rs * g.y + be.y;
    v.z = (v.z - mu) * rs * g.z + be.z;
    v.w = (v.w - mu) * rs * g.w + be.w;
    *(float4*)(out + (size_t)node * F + t * 4) = v;
}

// ---------------------------------------------------------------------------
extern "C" void kernel_launch(void* const* d_in, const int* in_sizes, int n_in,
                              void* d_out, int out_size, void* d_ws, size_t ws_size,
                              hipStream_t stream) {
    const float* x       = (const float*)d_in[0];
    const int*   ei      = (const int*)  d_in[1];
    const float* W       = (const float*)d_in[2];
    const float* att_src = (const float*)d_in[3];
    const float* att_dst = (const float*)d_in[4];
    const float* bias    = (const float*)d_in[5];
    const float* gamma   = (const float*)d_in[6];
    const float* beta    = (const float*)d_in[7];
    float*       out     = (float*)d_out;

    const int n  = in_sizes[0] / F;        // 50000
    const int E  = in_sizes[1] / 2;        // 800000
    const int ET = E + n;                  // edges + self loops

    // workspace layout (floats): xp | a_src | a_dst | maxv | sumv | Wf (~32 MB)
    float* xp    = (float*)d_ws;
    float* a_src = xp    + (size_t)n * F;
    float* a_dst = a_src + (size_t)n * NHEAD;
    float* maxv  = a_dst + (size_t)n * NHEAD;
    float* sumv  = maxv  + (size_t)n * NHEAD;
    float* Wf    = sumv  + (size_t)n * NHEAD;   // 16384 floats (64 KB)

    init_bufs   <<<(n * F + 255) / 256, 256, 0, stream>>>(maxv, sumv, out, n);
    pack_w      <<<(NHEAD * 32 * 32 + 255) / 256, 256, 0, stream>>>(W, Wf);
    gemm_xw     <<<(n + 15) / 16,       256, 0, stream>>>(x, Wf, xp, n);
    attn_scores <<<(n * NHEAD + 255) / 256, 256, 0, stream>>>(xp, att_src, att_dst,
                                                              a_src, a_dst, n);
    edge_max    <<<(ET + 255) / 256,    256, 0, stream>>>(ei, E, ET, a_src, a_dst, maxv);
    edge_sum    <<<(ET + 255) / 256,    256, 0, stream>>>(ei, E, ET, a_src, a_dst,
                                                          maxv, sumv);
    edge_scatter<<<(ET + 7) / 8,        256, 0, stream>>>(ei, E, ET, a_src, a_dst,
                                                          maxv, sumv, xp, out);
    layernorm   <<<(n + 7) / 8,         256, 0, stream>>>(out, bias, gamma, beta, n);
}